// GCN_4604204941855
// MI455X (gfx1250) — compile-verified
//
#include <hip/hip_runtime.h>

// Problem constants (match reference)
#define N_SRC0 991232
#define N_DST0 90112
#define N_E0   901120
#define N_DST1 8192
#define N_E1   81920
#define F_IN   128
#define H_DIM  256
#define C_DIM  47
#define C_PAD  48

typedef __attribute__((ext_vector_type(2))) float v2f;
typedef __attribute__((ext_vector_type(8))) float v8f;

// ---------------------------------------------------------------- utilities
__global__ void k_zero(float* __restrict__ p, int n) {
  int i = blockIdx.x * blockDim.x + threadIdx.x;
  if (i < n) p[i] = 0.0f;
}

__global__ void k_degree(const int* __restrict__ idx, int* deg, int ne) {
  int i = blockIdx.x * blockDim.x + threadIdx.x;
  if (i < ne) atomicAdd(&deg[idx[i]], 1);
}

// in-place: int degree -> rsqrt(max(deg,1))  (deg buffer reinterpreted)
__global__ void k_scale(int* degi, float* scale, int n) {
  int i = blockIdx.x * blockDim.x + threadIdx.x;
  if (i < n) {
    int c = degi[i];
    if (c < 1) c = 1;
    scale[i] = rsqrtf((float)c);
  }
}

// ----------------------------------------------- layer-1 edge scatter (128f)
// one wave32 per edge; lane l moves float4 [4l..4l+3] of the 128-float row
__global__ void __launch_bounds__(256)
k_edge_agg1(const float* __restrict__ x,
            const int* __restrict__ src,
            const int* __restrict__ dst,
            const float* __restrict__ scale_src,
            float* agg) {
  int wave = (int)((blockIdx.x * blockDim.x + threadIdx.x) >> 5);
  int lane = threadIdx.x & 31;
  if (wave >= N_E0) return;
  int s = src[wave];
  int d = dst[wave];
  float sc = scale_src[s];
  const float4* xr = reinterpret_cast<const float4*>(x + (size_t)s * F_IN);
  float4 v = xr[lane];
  float* o = agg + (size_t)d * F_IN + lane * 4;
  atomicAdd(o + 0, v.x * sc);
  atomicAdd(o + 1, v.y * sc);
  atomicAdd(o + 2, v.z * sc);
  atomicAdd(o + 3, v.w * sc);
}

// --------------------------------------- GEMM1: h = relu(s_in*(agg@W1) + b1)
// agg: 90112x128, W1: 128x256 row-major. One wave32 per 16x16 tile,
// K-loop of 32 x v_wmma_f32_16x16x4_f32.
__global__ void __launch_bounds__(256)
k_gemm1(const float* __restrict__ agg,
        const float* __restrict__ W1,
        const float* __restrict__ b1,
        const float* __restrict__ scale_dst,
        float* __restrict__ h) {
  int wave = (int)((blockIdx.x * blockDim.x + threadIdx.x) >> 5);
  int lane = threadIdx.x & 31;
  int tileM = wave >> 4;           // 16 N-tiles (256/16)
  int tileN = wave & 15;
  int m0 = tileM << 4, n0 = tileN << 4;
  int loc = lane & 15;             // m-index for A, n-index for B/D
  int khalf = lane >> 4;           // 0 -> K{0,1}, 1 -> K{2,3}

  const float* aRow = agg + (size_t)(m0 + loc) * F_IN + 2 * khalf;
  const float* bCol = W1 + (size_t)(2 * khalf) * H_DIM + n0 + loc;

  v8f acc = {};
#pragma unroll
  for (int k = 0; k < F_IN; k += 4) {
    v2f a = *reinterpret_cast<const v2f*>(aRow + k);
    v2f b;
    b.x = bCol[(size_t)k * H_DIM];
    b.y = bCol[(size_t)(k + 1) * H_DIM];
    acc = __builtin_amdgcn_wmma_f32_16x16x4_f32(false, a, false, b,
                                                (short)0, acc, false, false);
  }

  int n = n0 + loc;
  float bias = b1[n];
#pragma unroll
  for (int i = 0; i < 8; ++i) {
    int m = m0 + i + 8 * khalf;
    float v = acc[i] * scale_dst[m] + bias;
    v = v > 0.0f ? v : 0.0f;
    h[(size_t)m * H_DIM + n] = v;
  }
}

// ------------------------------------------- GEMM2: hw = (h*s_out) @ W2
// h: 90112x256, W2: 256x47 (N padded to 48 in hw). out_scale folded into A.
__global__ void __launch_bounds__(256)
k_gemm2(const float* __restrict__ h,
        const float* __restrict__ W2,
        const float* __restrict__ scale_src,
        float* __restrict__ hw) {
  int wave = (int)((blockIdx.x * blockDim.x + threadIdx.x) >> 5);
  int lane = threadIdx.x & 31;
  const int NT = 3;                // ceil(48/16)
  int tileM = wave / NT;
  int tileN = wave - tileM * NT;
  if (tileM >= N_DST0 / 16) return;
  int m0 = tileM << 4, n0 = tileN << 4;
  int loc = lane & 15;
  int khalf = lane >> 4;

  int mRow = m0 + loc;
  float asc = scale_src[mRow];
  const float* aRow = h + (size_t)mRow * H_DIM + 2 * khalf;
  int col = n0 + loc;
  bool bv = col < C_DIM;
  const float* bCol = W2 + (size_t)(2 * khalf) * C_DIM + (bv ? col : 0);

  v8f acc = {};
#pragma unroll 8
  for (int k = 0; k < H_DIM; k += 4) {
    v2f a = *reinterpret_cast<const v2f*>(aRow + k);
    a.x *= asc;
    a.y *= asc;
    v2f b;
    b.x = bv ? bCol[(size_t)k * C_DIM] : 0.0f;
    b.y = bv ? bCol[(size_t)(k + 1) * C_DIM] : 0.0f;
    acc = __builtin_amdgcn_wmma_f32_16x16x4_f32(false, a, false, b,
                                                (short)0, acc, false, false);
  }

#pragma unroll
  for (int i = 0; i < 8; ++i) {
    int m = m0 + i + 8 * khalf;
    hw[(size_t)m * C_PAD + (n0 + loc)] = acc[i];
  }
}

// ----------------------------------------------- layer-2 edge scatter (47f)
__global__ void __launch_bounds__(64)
k_edge_agg2(const float* __restrict__ hw,
            const int* __restrict__ src,
            const int* __restrict__ dst,
            float* out) {
  int e = blockIdx.x;
  int c = threadIdx.x;
  if (c >= C_DIM) return;
  int s = src[e];
  int d = dst[e];
  atomicAdd(&out[(size_t)d * C_DIM + c], hw[(size_t)s * C_PAD + c]);
}

// --------------------------------------------------- out = out*s_in + b2
__global__ void k_final(float* out,
                        const float* __restrict__ scale_dst,
                        const float* __restrict__ b2) {
  int i = blockIdx.x * blockDim.x + threadIdx.x;
  if (i >= N_DST1 * C_DIM) return;
  int r = i / C_DIM;
  int c = i - r * C_DIM;
  out[i] = out[i] * scale_dst[r] + b2[c];
}

extern "C" void kernel_launch(void* const* d_in, const int* in_sizes, int n_in,
                              void* d_out, int out_size, void* d_ws, size_t ws_size,
                              hipStream_t stream) {
  (void)in_sizes; (void)n_in; (void)ws_size;
  const float* x    = (const float*)d_in[0];
  const float* W1   = (const float*)d_in[1];
  const float* b1   = (const float*)d_in[2];
  const float* W2   = (const float*)d_in[3];
  const float* b2   = (const float*)d_in[4];
  const int*   src0 = (const int*)d_in[5];
  const int*   dst0 = (const int*)d_in[6];
  const int*   src1 = (const int*)d_in[7];
  const int*   dst1 = (const int*)d_in[8];
  float* out = (float*)d_out;

  // ---- workspace carve-up (~143 MB) ----
  char* ws = (char*)d_ws;
  const size_t DEG_CNT = (size_t)N_SRC0 + N_DST0 + N_DST0 + N_DST1;
  int*   degi       = (int*)ws;                 // int degrees, converted in place
  float* scale_src0 = (float*)ws;               // after k_scale: rsqrt(max(deg,1))
  float* scale_dst0 = scale_src0 + N_SRC0;
  float* scale_src1 = scale_dst0 + N_DST0;
  float* scale_dst1 = scale_src1 + N_DST0;
  size_t off = (DEG_CNT * 4 + 255) & ~(size_t)255;
  float* agg0 = (float*)(ws + off);             // 90112x128 f32
  float* hbuf = agg0 + (size_t)N_DST0 * F_IN;   // 90112x256 f32
  float* hw   = agg0;                           // reuse agg0 region after GEMM1

  int* deg_src0 = degi;
  int* deg_dst0 = degi + N_SRC0;
  int* deg_src1 = deg_dst0 + N_DST0;
  int* deg_dst1 = deg_src1 + N_DST0;

  auto cdiv = [](long a, long b) { return (int)((a + b - 1) / b); };

  // 1) zero degree region, agg0, and output accumulator
  k_zero<<<cdiv((long)DEG_CNT, 256), 256, 0, stream>>>((float*)degi, (int)DEG_CNT);
  k_zero<<<cdiv((long)N_DST0 * F_IN, 256), 256, 0, stream>>>(agg0, N_DST0 * F_IN);
  k_zero<<<cdiv(out_size, 256), 256, 0, stream>>>(out, out_size);

  // 2) degrees + in-place rsqrt scales
  k_degree<<<cdiv(N_E0, 256), 256, 0, stream>>>(src0, deg_src0, N_E0);
  k_degree<<<cdiv(N_E0, 256), 256, 0, stream>>>(dst0, deg_dst0, N_E0);
  k_degree<<<cdiv(N_E1, 256), 256, 0, stream>>>(src1, deg_src1, N_E1);
  k_degree<<<cdiv(N_E1, 256), 256, 0, stream>>>(dst1, deg_dst1, N_E1);
  k_scale<<<cdiv((long)DEG_CNT, 256), 256, 0, stream>>>(degi, (float*)degi, (int)DEG_CNT);

  // 3) layer-1 aggregate (wave32 per edge), then WMMA GEMM + bias + relu
  k_edge_agg1<<<N_E0 / 8, 256, 0, stream>>>(x, src0, dst0, scale_src0, agg0);
  k_gemm1<<<(N_DST0 / 16) * (H_DIM / 16) / 8, 256, 0, stream>>>(agg0, W1, b1, scale_dst0, hbuf);

  // 4) layer-2: multiply-first WMMA GEMM, scatter-add, finalize
  k_gemm2<<<cdiv((long)(N_DST0 / 16) * 3, 8), 256, 0, stream>>>(hbuf, W2, scale_src1, hw);
  k_edge_agg2<<<N_E1, 64, 0, stream>>>(hw, src1, dst1, out);
  k_final<<<cdiv(N_DST1 * C_DIM, 256), 256, 0, stream>>>(out, scale_dst1, b2);
}